// att_flow_layer_21912923144851
// MI455X (gfx1250) — compile-verified
//
#include <hip/hip_runtime.h>
#include <cstdint>
#include <cstddef>

#define NEGV (-1e30f)

typedef float v2f __attribute__((ext_vector_type(2)));
typedef float v8f __attribute__((ext_vector_type(8)));

#if defined(__HIP_DEVICE_COMPILE__) && defined(__gfx1250__)
#if !__has_builtin(__builtin_amdgcn_wmma_f32_16x16x4_f32)
#error "wmma f32 16x16x4 builtin missing on this toolchain"
#endif
#endif

#define LDS_S 68   // float stride: 68*4=272 bytes, multiple of 16 -> float4 aligned, conflict-friendly

// ---------------- Kernel 0: cvec = c@w_c + b_c ; qvec = q@w_q + b_q ----------------
__global__ __launch_bounds__(256) void k0_rowdots(
    const float* __restrict__ c, const float* __restrict__ q,
    const float* __restrict__ w_c, const float* __restrict__ b_c,
    const float* __restrict__ w_q, const float* __restrict__ b_q,
    float* __restrict__ cvec, float* __restrict__ qvec)
{
  const int wv = threadIdx.x >> 5;
  const int lane = threadIdx.x & 31;
  const int row = blockIdx.x * 8 + wv;           // 0 .. 9215
  const float* base; const float* w; float bias; float* dst;
  if (row < 16 * 512) {
    base = c + (size_t)row * 768; w = w_c; bias = *b_c; dst = cvec + row;
  } else {
    const int r2 = row - 16 * 512;               // 0 .. 1023
    base = q + (size_t)r2 * 768; w = w_q; bias = *b_q; dst = qvec + r2;
  }
  float s = 0.f;
  for (int d = lane; d < 768; d += 32) s += base[d] * w[d];
  #pragma unroll
  for (int m = 16; m >= 1; m >>= 1) s += __shfl_xor(s, m);
  if (lane == 0) *dst = s + bias;
}

// ---------------- Kernel 1: GEMM1 (s_cq) + softmax + GEMM2 (c2q) + out[0..3H) ----------------
__global__ __launch_bounds__(128) void k1_main(
    const float* __restrict__ c, const float* __restrict__ q,
    const float* __restrict__ c_mask, const float* __restrict__ q_mask,
    const float* __restrict__ w_cq, const float* __restrict__ b_cq_p,
    const float* __restrict__ cvec, const float* __restrict__ qvec,
    float* __restrict__ m_ws, float* __restrict__ out)
{
  __shared__ __align__(16) float cA[64 * LDS_S];  // c chunk (phase1: pre-scaled by w_cq; phase2: raw c)
  __shared__ __align__(16) float qB[64 * LDS_S];  // q chunk
  __shared__ __align__(16) float aS[64 * LDS_S];  // attention rows (A-matrix for GEMM2)

  const int b   = blockIdx.x >> 3;        // batch
  const int i0  = (blockIdx.x & 7) * 64;  // row tile base in CL
  const int wv  = threadIdx.x >> 5;       // wave 0..3 -> owns rows [wv*16, wv*16+16)
  const int lane = threadIdx.x & 31;
  const int lm   = lane & 15;
  const int koff = (lane >> 4) * 2;       // A/B K sub-offset for lane halves
  const int hi8  = (lane >> 4) * 8;       // C/D row offset for lane halves

  // ---------------- Phase 1: s_cq = (c*w_cq) @ q^T, K = 768 ----------------
  v8f acc[4];
  #pragma unroll
  for (int t = 0; t < 4; ++t)
    #pragma unroll
    for (int r = 0; r < 8; ++r) acc[t][r] = 0.f;

  for (int kk0 = 0; kk0 < 768; kk0 += 64) {
    __syncthreads();
    // stage 64x64 of c (scaled by w_cq) and q, float4 per thread x8
    #pragma unroll
    for (int s = 0; s < 8; ++s) {
      const int elem = (threadIdx.x + s * 128) * 4;
      const int row = elem >> 6, col = elem & 63;
      float4 wv4 = *(const float4*)(w_cq + kk0 + col);
      float4 cv4 = *(const float4*)(c + ((size_t)(b * 512 + i0 + row)) * 768 + kk0 + col);
      cv4.x *= wv4.x; cv4.y *= wv4.y; cv4.z *= wv4.z; cv4.w *= wv4.w;
      *(float4*)(&cA[row * LDS_S + col]) = cv4;
      float4 qv4 = *(const float4*)(q + ((size_t)(b * 64 + row)) * 768 + kk0 + col);
      *(float4*)(&qB[row * LDS_S + col]) = qv4;
    }
    __syncthreads();
    const float* ap = &cA[(wv * 16 + lm) * LDS_S];
    #pragma unroll 4
    for (int k = 0; k < 64; k += 4) {
      v2f A = *(const v2f*)(ap + k + koff);
      #pragma unroll
      for (int t = 0; t < 4; ++t) {
        v2f Bv = *(const v2f*)(&qB[(t * 16 + lm) * LDS_S + k + koff]);
        acc[t] = __builtin_amdgcn_wmma_f32_16x16x4_f32(
            false, A, false, Bv, (short)0, acc[t], false, false);
      }
    }
  }

  // ---------------- Epilogue: s = cq + cvec + qvec + b_cq; softmax over j ----------------
  const float bcq = *b_cq_p;
  float cvr[8];
  #pragma unroll
  for (int r = 0; r < 8; ++r) cvr[r] = cvec[b * 512 + i0 + wv * 16 + r + hi8];
  float qvr[4], qmr[4];
  #pragma unroll
  for (int t = 0; t < 4; ++t) {
    qvr[t] = qvec[b * 64 + t * 16 + lm];
    qmr[t] = q_mask[b * 64 + t * 16 + lm];
  }
  float sm[4][8], rawm[8], mskm[8];
  #pragma unroll
  for (int r = 0; r < 8; ++r) { rawm[r] = -3.0e38f; mskm[r] = -3.0e38f; }
  #pragma unroll
  for (int t = 0; t < 4; ++t)
    #pragma unroll
    for (int r = 0; r < 8; ++r) {
      float s = acc[t][r] + cvr[r] + qvr[t] + bcq;
      rawm[r] = fmaxf(rawm[r], s);
      float ms = s * qmr[t] + (1.f - qmr[t]) * NEGV;
      sm[t][r] = ms;
      mskm[r] = fmaxf(mskm[r], ms);
    }
  #pragma unroll
  for (int r = 0; r < 8; ++r)
    #pragma unroll
    for (int m = 8; m >= 1; m >>= 1) {   // reduce within 16-lane halves only
      rawm[r] = fmaxf(rawm[r], __shfl_xor(rawm[r], m));
      mskm[r] = fmaxf(mskm[r], __shfl_xor(mskm[r], m));
    }
  float den[8];
  #pragma unroll
  for (int r = 0; r < 8; ++r) den[r] = 0.f;
  #pragma unroll
  for (int t = 0; t < 4; ++t)
    #pragma unroll
    for (int r = 0; r < 8; ++r) {
      float e = __expf(sm[t][r] - mskm[r]);
      sm[t][r] = e; den[r] += e;
    }
  #pragma unroll
  for (int r = 0; r < 8; ++r)
    #pragma unroll
    for (int m = 8; m >= 1; m >>= 1) den[r] += __shfl_xor(den[r], m);
  #pragma unroll
  for (int t = 0; t < 4; ++t)
    #pragma unroll
    for (int r = 0; r < 8; ++r) sm[t][r] /= den[r];

  // row-max with c_mask for b_att (kernel 2)
  if (lm == 0) {
    #pragma unroll
    for (int r = 0; r < 8; ++r) {
      const int i = i0 + wv * 16 + r + hi8;
      const float cm = c_mask[b * 512 + i];
      m_ws[b * 512 + i] = rawm[r] * cm + (1.f - cm) * NEGV;
    }
  }
  // a -> LDS in A-matrix order (row = i, col = j)
  #pragma unroll
  for (int t = 0; t < 4; ++t)
    #pragma unroll
    for (int r = 0; r < 8; ++r)
      aS[(wv * 16 + r + hi8) * LDS_S + t * 16 + lm] = sm[t][r];

  // ---------------- Phase 2: c2q = a @ q, K = 64 (j), N = 768 (d), fused outputs ----------------
  for (int d0 = 0; d0 < 768; d0 += 64) {
    __syncthreads();
    #pragma unroll
    for (int s = 0; s < 8; ++s) {
      const int elem = (threadIdx.x + s * 128) * 4;
      const int row = elem >> 6, col = elem & 63;
      float4 cv4 = *(const float4*)(c + ((size_t)(b * 512 + i0 + row)) * 768 + d0 + col);
      *(float4*)(&cA[row * LDS_S + col]) = cv4;   // raw c chunk for output sections
      float4 qv4 = *(const float4*)(q + ((size_t)(b * 64 + row)) * 768 + d0 + col);
      *(float4*)(&qB[row * LDS_S + col]) = qv4;
    }
    __syncthreads();
    const float* ap2 = &aS[(wv * 16 + lm) * LDS_S];
    #pragma unroll
    for (int t = 0; t < 4; ++t) {
      v8f acc2;
      #pragma unroll
      for (int r = 0; r < 8; ++r) acc2[r] = 0.f;
      #pragma unroll 4
      for (int k = 0; k < 64; k += 4) {
        v2f A = *(const v2f*)(ap2 + k + koff);
        v2f Bv;
        Bv.x = qB[(k + koff) * LDS_S + t * 16 + lm];      // B[k][n] = q[j=k][d0+t*16+n]
        Bv.y = qB[(k + koff + 1) * LDS_S + t * 16 + lm];
        acc2 = __builtin_amdgcn_wmma_f32_16x16x4_f32(
            false, A, false, Bv, (short)0, acc2, false, false);
      }
      #pragma unroll
      for (int r = 0; r < 8; ++r) {
        const int rloc = wv * 16 + r + hi8;
        const float cval = cA[rloc * LDS_S + t * 16 + lm];
        const float c2 = acc2[r];
        const size_t ob = ((size_t)(b * 512 + i0 + rloc)) * 3072 + d0 + t * 16 + lm;
        out[ob]        = cval;        // section 0: c
        out[ob + 768]  = c2;          // section 1: c2q
        out[ob + 1536] = cval * c2;   // section 2: c * c2q
      }
    }
  }
}

// ---------------- Kernel 2: b_att softmax, q2c, out section 3 ----------------
__global__ __launch_bounds__(1024) void k2_q2c(
    const float* __restrict__ c, const float* __restrict__ m_ws,
    float* __restrict__ out)
{
  __shared__ float red[32];
  __shared__ float bm[512];
  __shared__ float q2c[768];
  const int b = blockIdx.x;
  const int tid = threadIdx.x;
  const int wv = tid >> 5, lane = tid & 31;

  // block max of m over i
  float mv = (tid < 512) ? m_ws[b * 512 + tid] : -3.0e38f;
  float v = mv;
  #pragma unroll
  for (int m = 16; m >= 1; m >>= 1) v = fmaxf(v, __shfl_xor(v, m));
  if (lane == 0) red[wv] = v;
  __syncthreads();
  if (tid < 32) {
    float t = red[tid];
    #pragma unroll
    for (int m = 16; m >= 1; m >>= 1) t = fmaxf(t, __shfl_xor(t, m));
    if (tid == 0) red[0] = t;
  }
  __syncthreads();
  const float bmax = red[0];
  __syncthreads();

  // block sum of exp
  float e = (tid < 512) ? __expf(mv - bmax) : 0.f;
  float sv = e;
  #pragma unroll
  for (int m = 16; m >= 1; m >>= 1) sv += __shfl_xor(sv, m);
  if (lane == 0) red[wv] = sv;
  __syncthreads();
  if (tid < 32) {
    float t = red[tid];
    #pragma unroll
    for (int m = 16; m >= 1; m >>= 1) t += __shfl_xor(t, m);
    if (tid == 0) red[0] = t;
  }
  __syncthreads();
  const float tot = red[0];
  if (tid < 512) bm[tid] = e / tot;
  __syncthreads();

  // q2c[d] = sum_i b_att[i] * c[b,i,d]
  if (tid < 768) {
    float a = 0.f;
    for (int i = 0; i < 512; ++i) a += bm[i] * c[((size_t)(b * 512 + i)) * 768 + tid];
    q2c[tid] = a;
  }
  __syncthreads();

  // out section 3: c * q2c (broadcast over i)
  for (int i = 0; i < 512; ++i) {
    for (int d = tid; d < 768; d += 1024) {
      const size_t ci = ((size_t)(b * 512 + i)) * 768 + d;
      out[((size_t)(b * 512 + i)) * 3072 + 2304 + d] = c[ci] * q2c[d];
    }
  }
}

extern "C" void kernel_launch(void* const* d_in, const int* in_sizes, int n_in,
                              void* d_out, int out_size, void* d_ws, size_t ws_size,
                              hipStream_t stream) {
  const float* c      = (const float*)d_in[0];
  const float* q      = (const float*)d_in[1];
  const float* c_mask = (const float*)d_in[2];
  const float* q_mask = (const float*)d_in[3];
  const float* w_c    = (const float*)d_in[4];
  const float* b_c    = (const float*)d_in[5];
  const float* w_q    = (const float*)d_in[6];
  const float* b_q    = (const float*)d_in[7];
  const float* w_cq   = (const float*)d_in[8];
  const float* b_cq   = (const float*)d_in[9];
  float* out = (float*)d_out;

  float* cvec = (float*)d_ws;            // 16*512
  float* qvec = cvec + 16 * 512;         // 16*64
  float* m_ws = qvec + 16 * 64;          // 16*512

  k0_rowdots<<<dim3((16 * 512 + 16 * 64) / 8), dim3(256), 0, stream>>>(
      c, q, w_c, b_c, w_q, b_q, cvec, qvec);
  k1_main<<<dim3(16 * 8), dim3(128), 0, stream>>>(
      c, q, c_mask, q_mask, w_cq, b_cq, cvec, qvec, m_ws, out);
  k2_q2c<<<dim3(16), dim3(1024), 0, stream>>>(c, m_ws, out);
}